// SortNet_77163382440845
// MI455X (gfx1250) — compile-verified
//
#include <hip/hip_runtime.h>

#define B_ 32
#define S_ 2048
#define E_ 6144
#define NEG_SLOPE 0.5f
#define LN_EPS 1e-5f
#define REN_EPS 2.220446049250313e-16f

typedef __attribute__((ext_vector_type(2))) float v2f;
typedef __attribute__((ext_vector_type(8))) float v8f;
typedef __attribute__((ext_vector_type(4))) unsigned int u32x4;
typedef __attribute__((ext_vector_type(8))) int i32x8;
typedef __attribute__((ext_vector_type(4))) int i32x4;

// ---------------- block reductions (256 threads, wave32) ----------------
__device__ __forceinline__ float wsum(float v) {
#pragma unroll
  for (int o = 16; o > 0; o >>= 1) v += __shfl_xor(v, o, 32);
  return v;
}
__device__ __forceinline__ float wmaxf(float v) {
#pragma unroll
  for (int o = 16; o > 0; o >>= 1) v = fmaxf(v, __shfl_xor(v, o, 32));
  return v;
}
__device__ __forceinline__ float wminf(float v) {
#pragma unroll
  for (int o = 16; o > 0; o >>= 1) v = fminf(v, __shfl_xor(v, o, 32));
  return v;
}
__device__ __forceinline__ float bsum(float v, float* sm) {
  int t = threadIdx.x;
  v = wsum(v);
  __syncthreads();
  if ((t & 31) == 0) sm[t >> 5] = v;
  __syncthreads();
  if (t == 0) { float r = 0.f; for (int i = 0; i < 8; ++i) r += sm[i]; sm[0] = r; }
  __syncthreads();
  return sm[0];
}
__device__ __forceinline__ float bmax(float v, float* sm) {
  int t = threadIdx.x;
  v = wmaxf(v);
  __syncthreads();
  if ((t & 31) == 0) sm[t >> 5] = v;
  __syncthreads();
  if (t == 0) { float r = sm[0]; for (int i = 1; i < 8; ++i) r = fmaxf(r, sm[i]); sm[0] = r; }
  __syncthreads();
  return sm[0];
}
__device__ __forceinline__ float bmin(float v, float* sm) {
  int t = threadIdx.x;
  v = wminf(v);
  __syncthreads();
  if ((t & 31) == 0) sm[t >> 5] = v;
  __syncthreads();
  if (t == 0) { float r = sm[0]; for (int i = 1; i < 8; ++i) r = fminf(r, sm[i]); sm[0] = r; }
  __syncthreads();
  return sm[0];
}

// ---------------- kernel 0: capture mean0/var0 (ddof=1), copy x -> xw ----
__global__ void k_init(const float* __restrict__ x, float* __restrict__ xw,
                       float* __restrict__ mean0, float* __restrict__ var0) {
  __shared__ float sm[8];
  int b = blockIdx.x, t = threadIdx.x;
  const float4* xr = (const float4*)(x + (size_t)b * S_);
  float4* xwr = (float4*)(xw + (size_t)b * S_);
  float s = 0.f, s2 = 0.f;
#pragma unroll
  for (int i = 0; i < 2; ++i) {
    float4 vv = xr[t + i * 256];
    xwr[t + i * 256] = vv;
    s += vv.x + vv.y + vv.z + vv.w;
    s2 += vv.x * vv.x + vv.y * vv.y + vv.z * vv.z + vv.w * vv.w;
  }
  s = bsum(s, sm);
  s2 = bsum(s2, sm);
  if (t == 0) {
    float mu = s / (float)S_;
    mean0[b] = mu;
    var0[b] = (s2 - (float)S_ * mu * mu) / (float)(S_ - 1);
  }
}

// ---------------- LN + qkv + k extrema (one block per batch row) --------
__global__ void k_ln_qkv(const float* __restrict__ xw, const float* __restrict__ g,
                         const float* __restrict__ bt, const float* __restrict__ ipw,
                         const float* __restrict__ ipb, float* __restrict__ q,
                         float* __restrict__ k, float* __restrict__ v,
                         float* __restrict__ kst) {
  __shared__ float sm[8];
  int b = blockIdx.x, t = threadIdx.x;
  const float* xr = xw + (size_t)b * S_;
  float xv[8];
  float s = 0.f, s2 = 0.f;
#pragma unroll
  for (int i = 0; i < 8; ++i) {
    xv[i] = xr[t + i * 256];
    s += xv[i];
    s2 += xv[i] * xv[i];
  }
  s = bsum(s, sm);
  s2 = bsum(s2, sm);
  float mu = s / (float)S_;
  float var = s2 / (float)S_ - mu * mu;  // biased, as nn.LayerNorm
  float inv = rsqrtf(var + LN_EPS);
  float w0 = ipw[0], w1 = ipw[1], w2 = ipw[2];
  float c0 = ipb[0], c1 = ipb[1], c2 = ipb[2];
  float kmx = -__builtin_inff(), kmn = __builtin_inff();
#pragma unroll
  for (int i = 0; i < 8; ++i) {
    int idx = t + i * 256;
    float u = (xv[i] - mu) * inv * g[idx] + bt[idx];
    float qq = u * w0 + c0;
    float kk = u * w1 + c1;
    float vv = u * w2 + c2;
    q[(size_t)b * S_ + idx] = qq;
    k[(size_t)b * S_ + idx] = kk;
    v[(size_t)b * S_ + idx] = vv;
    kmx = fmaxf(kmx, kk);
    kmn = fminf(kmn, kk);
  }
  kmx = bmax(kmx, sm);
  kmn = bmin(kmn, sm);
  if (t == 0) { kst[2 * b] = kmx; kst[2 * b + 1] = kmn; }
}

// ---------------- rank-1 attention: streaming softmax over j ------------
__global__ void k_attn(const float* __restrict__ q, const float* __restrict__ k,
                       const float* __restrict__ v, const float* __restrict__ kst,
                       const float* __restrict__ owp, const float* __restrict__ obp,
                       float* __restrict__ xw) {
  __shared__ float lk[S_];
  __shared__ float lv[S_];
  int b = blockIdx.x >> 3;
  int ic = blockIdx.x & 7;
  int t = threadIdx.x;
  const float4* kr = (const float4*)(k + (size_t)b * S_);
  const float4* vr = (const float4*)(v + (size_t)b * S_);
#pragma unroll
  for (int i = 0; i < 2; ++i) {
    ((float4*)lk)[t + i * 256] = kr[t + i * 256];
    ((float4*)lv)[t + i * 256] = vr[t + i * 256];
  }
  __syncthreads();
  int i = ic * 256 + t;
  float qi = q[(size_t)b * S_ + i];
  float km = kst[2 * b], kn = kst[2 * b + 1];
  float mi = fmaxf(qi * km, qi * kn);  // exact row max of q_i * k_j
  float Z = 0.f, acc = 0.f;
  const float4* lk4 = (const float4*)lk;
  const float4* lv4 = (const float4*)lv;
  for (int j = 0; j < S_ / 4; ++j) {
    float4 kk = lk4[j];
    float4 vv = lv4[j];
    float e;
    e = __expf(qi * kk.x - mi); Z += e; acc += e * vv.x;
    e = __expf(qi * kk.y - mi); Z += e; acc += e * vv.y;
    e = __expf(qi * kk.z - mi); Z += e; acc += e * vv.z;
    e = __expf(qi * kk.w - mi); Z += e; acc += e * vv.w;
  }
  float o = (acc / Z) * owp[0] + obp[0];
  xw[(size_t)b * S_ + i] += o;  // residual, in place (this thread owns the element)
}

// ---------------- LN only (MLP input) -----------------------------------
__global__ void k_ln(const float* __restrict__ xw, const float* __restrict__ g,
                     const float* __restrict__ bt, float* __restrict__ u) {
  __shared__ float sm[8];
  int b = blockIdx.x, t = threadIdx.x;
  const float* xr = xw + (size_t)b * S_;
  float xv[8];
  float s = 0.f, s2 = 0.f;
#pragma unroll
  for (int i = 0; i < 8; ++i) {
    xv[i] = xr[t + i * 256];
    s += xv[i];
    s2 += xv[i] * xv[i];
  }
  s = bsum(s, sm);
  s2 = bsum(s2, sm);
  float mu = s / (float)S_;
  float var = s2 / (float)S_ - mu * mu;
  float inv = rsqrtf(var + LN_EPS);
#pragma unroll
  for (int i = 0; i < 8; ++i) {
    int idx = t + i * 256;
    u[(size_t)b * S_ + idx] = (xv[i] - mu) * inv * g[idx] + bt[idx];
  }
}

// ---------------- TDM: 2D tile (rows x KC cols, row stride K) -> LDS -----
// D# per CDNA5 ISA. Pad 4 DWORDs every 128 DWORDs to
// reproduce the LDS row stride KC+PAD (bank-conflict-free ds reads).
#define KC 128
#define PAD 4
#define ROWS_W 64
#define ROWS_A 32
#define LDS_STRIDE (KC + PAD)
#define BUF_FLOATS ((ROWS_W + ROWS_A) * LDS_STRIDE)

__device__ __forceinline__ void tdm_load_2d(unsigned int lds_byte, const float* gptr,
                                            unsigned int rows, unsigned int K) {
  unsigned long long ga = (unsigned long long)(uintptr_t)gptr;
  u32x4 g0;
  g0[0] = 1u;                                        // count=1, user descriptor
  g0[1] = lds_byte;                                  // lds_addr (bytes)
  g0[2] = (unsigned int)(ga & 0xFFFFFFFFu);          // global_addr[31:0]
  g0[3] = (unsigned int)((ga >> 32) & 0x1FFFFFFu) | (2u << 30);  // addr[56:32]|type=2
  i32x8 g1;
  // workgroup_mask=0 | data_size=2 (4B) | pad_enable=1 | pad_interval=6 (128 dw)
  // | pad_amount=3 (4 dwords)
  g1[0] = (int)((2u << 16) | (1u << 20) | (6u << 22) | (3u << 25));
  g1[1] = (int)((K & 0xFFFFu) << 16);                // tensor_dim0[15:0]
  g1[2] = (int)(((K >> 16) & 0xFFFFu) | ((rows & 0xFFFFu) << 16)); // dim0 hi|dim1 lo
  g1[3] = (int)((unsigned int)KC << 16);             // tensor_dim1 hi=0 | tile_dim0=KC
  g1[4] = (int)(rows & 0xFFFFu);                     // tile_dim1=rows, tile_dim2=0
  g1[5] = (int)K;                                    // tensor_dim0_stride[31:0] = K
  g1[6] = 0;                                        // stride hi | tensor_dim1_stride lo
  g1[7] = 0;
  i32x4 z4 = {0, 0, 0, 0};
#if defined(__clang_major__) && (__clang_major__ >= 23)
  i32x8 z8 = {0, 0, 0, 0, 0, 0, 0, 0};
  __builtin_amdgcn_tensor_load_to_lds(g0, g1, z4, z4, z8, 0);
#else
  __builtin_amdgcn_tensor_load_to_lds(g0, g1, z4, z4, 0);
#endif
}

// ---------------- WMMA f32 GEMM: Out[32,Nout] = Act[32,K] @ W[Nout,K]^T --
// C^T tiles: D(e x m) = W_tile(16x4) x Act^T_tile(4x16). 128 threads = 4
// waves; wave owns 16 e-rows, two accumulators (m 0..15 / 16..31). Both W
// and Act chunks are staged to LDS by the Tensor Data Mover, double-buffered
// (TENSORcnt pipelining), so the inner loop is pure ds_load + v_wmma.
// NOTE: no pointer arrays with LDS-derived initializers (ld.lld cannot
// materialize addrspacecast(@lds) in static initializers) — all buffer
// addresses are runtime integer offsets off one lds_base.
__global__ void k_gemm(const float* __restrict__ W, const float* __restrict__ Act,
                       const float* __restrict__ bias, const float* __restrict__ resid,
                       float* __restrict__ Out, int K, int Nout, int mode) {
  extern __shared__ float lds[];
  const int t = threadIdx.x;
  const int lane = t & 31;
  const int wave = t >> 5;                 // 0..3
  const int e0 = blockIdx.x * 64 + wave * 16;
  const int mrow = lane & 15;              // A: M = lane&15 ; B/D: N = lane&15
  const int khalf = (lane >> 4) << 1;      // lanes 0-15 -> K{0,1}; 16-31 -> K{2,3}

  const unsigned int lds_base = (unsigned int)(uintptr_t)(void*)&lds[0];
  const unsigned int aoff = ROWS_W * LDS_STRIDE;   // Act region offset (floats)
  const float* wblk = W + (size_t)blockIdx.x * 64 * K;
  const int nchunk = K / KC;

  if (wave == 0) {  // prologue: chunk 0 into buffer 0 (TDM, wave-level issue)
    tdm_load_2d(lds_base, wblk, ROWS_W, (unsigned int)K);
    tdm_load_2d(lds_base + aoff * 4u, Act, ROWS_A, (unsigned int)K);
  }

  v8f c0 = {};
  v8f c1 = {};
  for (int c = 0; c < nchunk; ++c) {
    const unsigned int cur = (unsigned int)(c & 1) * BUF_FLOATS;  // float offset
    if (wave == 0) {
      if (c + 1 < nchunk) {
        const unsigned int nxt = (unsigned int)((c + 1) & 1) * BUF_FLOATS;
        tdm_load_2d(lds_base + nxt * 4u, wblk + (size_t)(c + 1) * KC,
                    ROWS_W, (unsigned int)K);
        tdm_load_2d(lds_base + (nxt + aoff) * 4u, Act + (size_t)(c + 1) * KC,
                    ROWS_A, (unsigned int)K);
        __builtin_amdgcn_s_wait_tensorcnt(2);  // chunk c landed (TDM is in-order)
      } else {
        __builtin_amdgcn_s_wait_tensorcnt(0);
      }
    }
    __syncthreads();  // chunk c visible to all waves
    const float* lw  = lds + cur + (unsigned int)(wave * 16 + mrow) * LDS_STRIDE + khalf;
    const float* la0 = lds + cur + aoff + (unsigned int)mrow * LDS_STRIDE + khalf;
    const float* la1 = lds + cur + aoff + (unsigned int)(mrow + 16) * LDS_STRIDE + khalf;
#pragma unroll 4
    for (int kk = 0; kk < KC; kk += 4) {
      v2f a  = *(const v2f*)(lw + kk);    // A 16x4 f32 tile
      v2f b0 = *(const v2f*)(la0 + kk);   // B 4x16, n = m 0..15
      v2f b1 = *(const v2f*)(la1 + kk);   // B 4x16, n = m 16..31
      c0 = __builtin_amdgcn_wmma_f32_16x16x4_f32(false, a, false, b0,
                                                 (short)0, c0, false, false);
      c1 = __builtin_amdgcn_wmma_f32_16x16x4_f32(false, a, false, b1,
                                                 (short)0, c1, false, false);
    }
    __syncthreads();  // all waves done with buf[cur] before TDM overwrites it
  }

  // D layout: VGPR r -> M = r (lanes 0-15) / r+8 (lanes 16-31), N = lane&15
  const int erow = e0 + ((lane >> 4) << 3);
  const int n = mrow;
#pragma unroll
  for (int r = 0; r < 8; ++r) {
    int e = erow + r;
    float bb = bias[e];
    float o0 = c0[r] + bb;
    float o1 = c1[r] + bb;
    if (mode == 0) {  // GEMM1: LeakyReLU(0.5)
      o0 = (o0 >= 0.f) ? o0 : NEG_SLOPE * o0;
      o1 = (o1 >= 0.f) ? o1 : NEG_SLOPE * o1;
    } else {          // GEMM2: + residual
      o0 += resid[(size_t)n * Nout + e];
      o1 += resid[(size_t)(n + 16) * Nout + e];
    }
    Out[(size_t)n * Nout + e] = o0;
    Out[(size_t)(n + 16) * Nout + e] = o1;
  }
}

// ---------------- final renorm ------------------------------------------
__global__ void k_final(const float* __restrict__ xw, const float* __restrict__ mean0,
                        const float* __restrict__ var0, float* __restrict__ out) {
  __shared__ float sm[8];
  int b = blockIdx.x, t = threadIdx.x;
  const float* xr = xw + (size_t)b * S_;
  float xv[8];
  float s = 0.f, s2 = 0.f;
#pragma unroll
  for (int i = 0; i < 8; ++i) {
    xv[i] = xr[t + i * 256];
    s += xv[i];
    s2 += xv[i] * xv[i];
  }
  s = bsum(s, sm);
  s2 = bsum(s2, sm);
  float mu = s / (float)S_;
  float var = (s2 - (float)S_ * mu * mu) / (float)(S_ - 1);  // ddof=1
  float scale = sqrtf(var0[b] + REN_EPS) / sqrtf(var + REN_EPS);
  float m0 = mean0[b];
#pragma unroll
  for (int i = 0; i < 8; ++i) {
    out[(size_t)b * S_ + t + i * 256] = (xv[i] - mu) * scale + m0;
  }
}

// ---------------- host orchestration ------------------------------------
extern "C" void kernel_launch(void* const* d_in, const int* in_sizes, int n_in,
                              void* d_out, int out_size, void* d_ws, size_t ws_size,
                              hipStream_t stream) {
  (void)in_sizes; (void)n_in; (void)out_size; (void)ws_size;
  const float* x   = (const float*)d_in[0];
  const float* ag  = (const float*)d_in[1];
  const float* ab  = (const float*)d_in[2];
  const float* ipw = (const float*)d_in[3];
  const float* ipb = (const float*)d_in[4];
  const float* ow  = (const float*)d_in[5];
  const float* ob  = (const float*)d_in[6];
  const float* mg  = (const float*)d_in[7];
  const float* mb  = (const float*)d_in[8];
  const float* W1  = (const float*)d_in[9];
  const float* b1  = (const float*)d_in[10];
  const float* W2  = (const float*)d_in[11];
  const float* b2  = (const float*)d_in[12];
  float* out = (float*)d_out;

  float* ws = (float*)d_ws;
  float* xw = ws;                          // B*S
  float* u  = xw + (size_t)B_ * S_;        // B*S
  float* q  = u  + (size_t)B_ * S_;        // B*S
  float* k  = q  + (size_t)B_ * S_;        // B*S
  float* v  = k  + (size_t)B_ * S_;        // B*S
  float* H  = v  + (size_t)B_ * S_;        // B*E
  float* st = H  + (size_t)B_ * E_;        // mean0[32], var0[32], kst[64]

  const size_t gemm_lds = (size_t)2 * BUF_FLOATS * sizeof(float);  // ~101 KB

  k_init<<<B_, 256, 0, stream>>>(x, xw, st, st + 32);
  for (int L = 0; L < 2; ++L) {
    k_ln_qkv<<<B_, 256, 0, stream>>>(xw, ag + L * S_, ab + L * S_,
                                     ipw + L * 3, ipb + L * 3, q, k, v, st + 64);
    k_attn<<<B_ * 8, 256, 0, stream>>>(q, k, v, st + 64, ow + L, ob + L, xw);
    k_ln<<<B_, 256, 0, stream>>>(xw, mg + L * S_, mb + L * S_, u);
    k_gemm<<<E_ / 64, 128, gemm_lds, stream>>>(W1 + (size_t)L * E_ * S_, u,
                                               b1 + (size_t)L * E_, nullptr,
                                               H, S_, E_, 0);
    k_gemm<<<S_ / 64, 128, gemm_lds, stream>>>(W2 + (size_t)L * S_ * E_, H,
                                               b2 + (size_t)L * S_, xw,
                                               xw, E_, S_, 1);
  }
  k_final<<<B_, 256, 0, stream>>>(xw, st, st + 32, out);
}